// ContextualLoss_38577396253221
// MI455X (gfx1250) — compile-verified
//
#include <hip/hip_runtime.h>
#include <math.h>

// CDNA5 / gfx1250 contextual-loss kernel.
// Core op: V_WMMA_F32_16X16X4_F32 computing 16x16 similarity tiles with K=4
// (C=2 channels + 2 zero-pad rows). Inputs/outputs stay f32 end-to-end.
// Staging uses GLOBAL_LOAD_ASYNC_TO_LDS_B128 (ASYNCcnt) when available.

typedef __attribute__((ext_vector_type(2))) float v2f;
typedef __attribute__((ext_vector_type(8))) float v8f;
typedef __attribute__((ext_vector_type(4))) int   v4i;

#define AS1 __attribute__((address_space(1)))
#define AS3 __attribute__((address_space(3)))

#define NPIX  65536            // H*W
#define NIMG  8                // B
#define WAVES 8                // waves per block (wave32) -> 128 pred rows/block
#define CHUNK_PIX 2048         // target pixels staged in LDS per stage (16 KB)
#define STAGES (NPIX / CHUNK_PIX)          // 32
#define TILES_PER_STAGE (CHUNK_PIX / 16)   // 128
#define GROUP 8                // B-tiles loaded per ds clause

// ---------------------------------------------------------------------------
// Kernel 1: channel-normalize [B,2,N] f32 -> [B,N] float2 (unit 2-vectors)
// ---------------------------------------------------------------------------
__global__ void cx_normalize(const float* __restrict__ src,
                             float2* __restrict__ dst) {
  int i = blockIdx.x * blockDim.x + threadIdx.x;  // over NIMG*NPIX
  int b = i >> 16;
  int n = i & (NPIX - 1);
  const float* p = src + (size_t)b * 2 * NPIX;
  float x = p[n];
  float y = p[n + NPIX];
  float r = sqrtf(x * x + y * y);
  float inv = 1.0f / fmaxf(r, 1e-8f);
  dst[i] = make_float2(x * inv, y * inv);
}

// ---------------------------------------------------------------------------
// Kernel 2: zero the per-image accumulators (graph-capture safe, no memset)
// ---------------------------------------------------------------------------
__global__ void cx_zero(float* __restrict__ s) {
  if (threadIdx.x < NIMG) s[threadIdx.x] = 0.0f;
}

// ---------------------------------------------------------------------------
// Kernel 3: main WMMA loop.
//   grid = (NPIX/(16*WAVES), NIMG), block = 256 threads (8 wave32 waves).
//   Each wave owns one 16-row pred A-fragment for the whole kernel and
//   streams all target tiles through LDS, accumulating a running row-max
//   directly in the WMMA D-register layout.
// ---------------------------------------------------------------------------
__global__ void __launch_bounds__(256)
cx_main(const float2* __restrict__ pn, const float2* __restrict__ tn,
        float* __restrict__ imgsum) {
  // 16 KB stage + 128 B zero pad (read by lanes 16-31 = K=2,3 rows of B).
  __shared__ __align__(16) float2 stage[CHUNK_PIX + 16];

  const int b    = blockIdx.y;
  const int wave = threadIdx.x >> 5;
  const int lane = threadIdx.x & 31;

  const float2* pimg = pn + (size_t)b * NPIX;
  const float2* timg = tn + (size_t)b * NPIX;

  // Zero pad written once; covered by the first stage barrier.
  if (threadIdx.x < 16) stage[CHUNK_PIX + threadIdx.x] = make_float2(0.f, 0.f);

  // A fragment (16x4 f32): lanes 0-15 hold row m0+lane with K=0,1 in the two
  // VGPRs; lanes 16-31 hold K=2,3 which are the zero padding for C=2.
  const int m0 = (blockIdx.x * WAVES + wave) * 16;
  v2f afrag = {0.0f, 0.0f};
  if (lane < 16) {
    float2 p = pimg[m0 + lane];
    afrag[0] = p.x;
    afrag[1] = p.y;
  }

  // Per-lane unconditional B pointer: lanes 0-15 walk the staged columns,
  // lanes 16-31 always re-read the zero pad. No EXEC manipulation needed.
  float2* bbase;
  int bstride;
  if (lane < 16) { bbase = &stage[lane];                    bstride = 16; }
  else           { bbase = &stage[CHUNK_PIX + (lane - 16)]; bstride = 0;  }

  // Running max in D layout: VGPR r, lanes 0-15 => row m0+r, lanes 16-31 =>
  // row m0+8+r; lane's column subset is {16*jt + (lane&15)} over all tiles.
  v8f rowmax;
#pragma unroll
  for (int r = 0; r < 8; ++r) rowmax[r] = -2.0f;

  for (int s = 0; s < STAGES; ++s) {
    // ---- Stage CHUNK_PIX float2 (16 KB) global -> LDS -------------------
    const float4* gsrc = (const float4*)(timg + s * CHUNK_PIX);
    float4* ldst = (float4*)stage;
#if __has_builtin(__builtin_amdgcn_global_load_async_to_lds_b128)
#pragma unroll
    for (int k = 0; k < (CHUNK_PIX * 2 / 4) / 256; ++k) {
      __builtin_amdgcn_global_load_async_to_lds_b128(
          (AS1 v4i*)(gsrc + threadIdx.x + k * 256),
          (AS3 v4i*)(ldst + threadIdx.x + k * 256),
          0, 0);
    }
#if __has_builtin(__builtin_amdgcn_s_wait_asynccnt)
    __builtin_amdgcn_s_wait_asynccnt(0);
#else
    asm volatile("s_wait_asynccnt 0x0" ::: "memory");
#endif
#else
#pragma unroll
    for (int k = 0; k < (CHUNK_PIX * 2 / 4) / 256; ++k)
      ldst[threadIdx.x + k * 256] = gsrc[threadIdx.x + k * 256];
#endif
    __syncthreads();

    // ---- WMMA over the staged tiles, GROUP tiles per ds clause ----------
    float2* bp = bbase;
    for (int jt = 0; jt < TILES_PER_STAGE; jt += GROUP) {
      float2 t[GROUP];
#pragma unroll
      for (int u = 0; u < GROUP; ++u) t[u] = bp[u * bstride];
      bp += GROUP * bstride;
#pragma unroll
      for (int u = 0; u < GROUP; ++u) {
        v2f bfrag;
        bfrag[0] = t[u].x;
        bfrag[1] = t[u].y;
        v8f c = {};
        // D = A(16x4) x B(4x16) + 0 -> v_wmma_f32_16x16x4_f32
        v8f sim = __builtin_amdgcn_wmma_f32_16x16x4_f32(
            /*neg_a=*/false, afrag, /*neg_b=*/false, bfrag,
            /*c_mod=*/(short)0, c, /*reuse_a=*/false, /*reuse_b=*/false);
#pragma unroll
        for (int r = 0; r < 8; ++r) rowmax[r] = fmaxf(rowmax[r], sim[r]);
      }
    }
    __syncthreads();
  }

  // Reduce max across the 16 lanes of each half (columns), per row register.
#pragma unroll
  for (int r = 0; r < 8; ++r) {
    float v = rowmax[r];
    v = fmaxf(v, __shfl_xor(v, 1, 32));
    v = fmaxf(v, __shfl_xor(v, 2, 32));
    v = fmaxf(v, __shfl_xor(v, 4, 32));
    v = fmaxf(v, __shfl_xor(v, 8, 32));
    rowmax[r] = v;
  }

  // clip(max(sim),1e-6,1) == clip(max(clip(sim,-1,1)),1e-6,1): clamp once.
  if (lane == 0 || lane == 16) {
    float acc = 0.0f;
#pragma unroll
    for (int r = 0; r < 8; ++r)
      acc += fminf(fmaxf(rowmax[r], 1e-6f), 1.0f);
    atomicAdd(&imgsum[b], acc);  // global_atomic_add_f32
  }
}

// ---------------------------------------------------------------------------
// Kernel 4: loss = mean_b( -log( imgsum[b] / N ) )
// ---------------------------------------------------------------------------
__global__ void cx_final(const float* __restrict__ imgsum,
                         float* __restrict__ out) {
  if (threadIdx.x == 0) {
    float acc = 0.0f;
#pragma unroll
    for (int b = 0; b < NIMG; ++b)
      acc += -logf(imgsum[b] * (1.0f / (float)NPIX));
    out[0] = acc * (1.0f / (float)NIMG);
  }
}

// ---------------------------------------------------------------------------
extern "C" void kernel_launch(void* const* d_in, const int* in_sizes, int n_in,
                              void* d_out, int out_size, void* d_ws,
                              size_t ws_size, hipStream_t stream) {
  const float* pred   = (const float*)d_in[0];
  const float* target = (const float*)d_in[1];

  // Workspace layout: pn [8*65536 float2] | tn [8*65536 float2] | imgsum[8]
  float2* pn = (float2*)d_ws;
  float2* tn = pn + (size_t)NIMG * NPIX;
  float* imgsum = (float*)(tn + (size_t)NIMG * NPIX);

  cx_normalize<<<(NIMG * NPIX) / 256, 256, 0, stream>>>(pred, pn);
  cx_normalize<<<(NIMG * NPIX) / 256, 256, 0, stream>>>(target, tn);
  cx_zero<<<1, 32, 0, stream>>>(imgsum);

  dim3 grid(NPIX / (16 * WAVES), NIMG);
  cx_main<<<grid, 256, 0, stream>>>(pn, tn, imgsum);

  cx_final<<<1, 32, 0, stream>>>(imgsum, (float*)d_out);
}